// GraphReasoningModel_67293547594153
// MI455X (gfx1250) — compile-verified
//
#include <hip/hip_runtime.h>
#include <hip/hip_bf16.h>
#include <math.h>

#define BB 16
#define LL 64
#define DD 768
#define EE 100000
#define RR 200
#define TT 1000000
#define NWAYS 2
#define NSTEPS 3

typedef __attribute__((ext_vector_type(2))) float v2f;
typedef __attribute__((ext_vector_type(8))) float v8f;

__device__ __forceinline__ void atomAddF(float* p, float v) {
#if defined(__HIP_DEVICE_COMPILE__)
  unsafeAtomicAdd(p, v);   // native global_atomic_add_f32 (L2 atomic units)
#else
  (void)p; (void)v;
#endif
}

// ---------------------------------------------------------------- init / zero
__global__ void k_init(float* __restrict__ out, float* __restrict__ acc0,
                       float* __restrict__ red) {
  int idx = blockIdx.x * blockDim.x + threadIdx.x;
  if (idx < BB * EE) { out[idx] = 0.f; acc0[idx] = 0.f; }
  if (idx < 64) red[idx] = 0.f;
}

__global__ void k_fzero(float* __restrict__ nxt, float* __restrict__ red) {
  int idx = blockIdx.x * blockDim.x + threadIdx.x;
  if (idx < BB * EE) nxt[idx] = 0.f;
  if (idx < 64) red[idx] = 0.f;
}

// -------------------------------------------------- qem = q @ match_w + b
__global__ void k_qem(const float* __restrict__ q, const float* __restrict__ w,
                      const float* __restrict__ bias, float* __restrict__ out) {
  int j = blockIdx.x * blockDim.x + threadIdx.x;
  if (j >= DD) return;
  float acc[BB];
#pragma unroll
  for (int b = 0; b < BB; ++b) acc[b] = 0.f;
  for (int d = 0; d < DD; ++d) {
    float wv = w[d * DD + j];
#pragma unroll
    for (int b = 0; b < BB; ++b) acc[b] += q[b * DD + d] * wv;
  }
#pragma unroll
  for (int b = 0; b < BB; ++b) out[b * DD + j] = acc[b] + bias[j];
}

// ------------------------------ cq[i] = tanh(q @ step_w[i] + step_b[i]), i<6
__global__ void k_cq(const float* __restrict__ q, const float* __restrict__ sw,
                     const float* __restrict__ sb, float* __restrict__ cq_all) {
  int j = blockIdx.x * blockDim.x + threadIdx.x;
  int i = blockIdx.y;
  if (j >= DD) return;
  const float* w = sw + (size_t)i * DD * DD;
  float acc[BB];
#pragma unroll
  for (int b = 0; b < BB; ++b) acc[b] = 0.f;
  for (int d = 0; d < DD; ++d) {
    float wv = w[d * DD + j];
#pragma unroll
    for (int b = 0; b < BB; ++b) acc[b] += q[b * DD + d] * wv;
  }
#pragma unroll
  for (int b = 0; b < BB; ++b)
    cq_all[((size_t)i * BB + b) * DD + j] = tanhf(acc[b] + sb[i * DD + j]);
}

// -------------------------------------- hop_attn[w,b,:] = softmax(q@hop_w+b)
__global__ void k_hop(const float* __restrict__ q, const float* __restrict__ hw,
                      const float* __restrict__ hb, float* __restrict__ hop_attn) {
  int tid = threadIdx.x;
  if (tid >= NWAYS * BB) return;
  int b = tid & 15, w = tid >> 4;
  float lg[NSTEPS];
#pragma unroll
  for (int s = 0; s < NSTEPS; ++s) lg[s] = hb[w * NSTEPS + s];
  for (int d = 0; d < DD; ++d) {
    float qv = q[b * DD + d];
#pragma unroll
    for (int s = 0; s < NSTEPS; ++s) lg[s] += qv * hw[((size_t)w * DD + d) * NSTEPS + s];
  }
  float mx = fmaxf(lg[0], fmaxf(lg[1], lg[2]));
  float e0 = __expf(lg[0] - mx), e1 = __expf(lg[1] - mx), e2 = __expf(lg[2] - mx);
  float inv = 1.f / (e0 + e1 + e2);
  hop_attn[(w * BB + b) * NSTEPS + 0] = e0 * inv;
  hop_attn[(w * BB + b) * NSTEPS + 1] = e1 * inv;
  hop_attn[(w * BB + b) * NSTEPS + 2] = e2 * inv;
}

// ---------- per (i,b): word attention -> ctx -> rel softmax -> wr (pre-scaled)
__global__ void k_ctx_rel(const float* __restrict__ cq_all, const float* __restrict__ qwh,
                          const int* __restrict__ amask, const float* __restrict__ rel_w,
                          const float* __restrict__ rel_b, const float* __restrict__ rimp,
                          const float* __restrict__ temp, float* __restrict__ wr_all) {
  int i = blockIdx.x, b = blockIdx.y, w = i / NSTEPS;
  int tid = threadIdx.x;
  __shared__ float slg[LL];
  __shared__ float sctx[DD];
  __shared__ float srl[RR];
  const float* cq = cq_all + ((size_t)i * BB + b) * DD;
  if (tid < LL) {
    float acc = 0.f;
    const float* row = qwh + ((size_t)b * LL + tid) * DD;
    for (int d = 0; d < DD; ++d) acc += cq[d] * row[d];
    slg[tid] = acc / temp[0];
  }
  __syncthreads();
  if (tid == 0) {
    float mx = -1e30f;
    for (int l = 0; l < LL; ++l) mx = fmaxf(mx, slg[l]);
    float ssum = 0.f;
    for (int l = 0; l < LL; ++l) { float e = __expf(slg[l] - mx); slg[l] = e; ssum += e; }
    float s2 = 0.f;
    for (int l = 0; l < LL; ++l) {
      float v = (slg[l] / ssum) * (float)amask[b * LL + l];
      slg[l] = v; s2 += v;
    }
    float inv = 1.f / (s2 + 1e-6f);
    for (int l = 0; l < LL; ++l) slg[l] *= inv;
  }
  __syncthreads();
  for (int d = tid; d < DD; d += blockDim.x) {
    float acc = 0.f;
    for (int l = 0; l < LL; ++l) acc += slg[l] * qwh[((size_t)b * LL + l) * DD + d];
    sctx[d] = acc;
  }
  __syncthreads();
  if (tid < RR) {
    float acc = rel_b[w * RR + tid];
    for (int d = 0; d < DD; ++d) acc += sctx[d] * rel_w[((size_t)w * DD + d) * RR + tid];
    srl[tid] = acc;
  }
  __syncthreads();
  if (tid == 0) {
    float mx = -1e30f;
    for (int r = 0; r < RR; ++r) mx = fmaxf(mx, srl[r]);
    float ssum = 0.f;
    for (int r = 0; r < RR; ++r) { float e = __expf(srl[r] - mx); srl[r] = e; ssum += e; }
    float inv = 1.f / ssum;
    for (int r = 0; r < RR; ++r) srl[r] *= inv;
  }
  __syncthreads();
  if (tid < RR)
    wr_all[((size_t)i * BB + b) * RR + tid] = srl[tid] * rimp[tid];
}

// --------------- direct = sigmoid(qem @ ent^T / sqrt(D)) via WMMA f32 16x16x4
// A (16xK) staged in LDS with padded stride 772 to avoid 16-way bank conflicts.
__global__ void k_direct(const float* __restrict__ qem, const float* __restrict__ ent,
                         float* __restrict__ direct) {
  __shared__ float sq[16 * 772];
  for (int idx = threadIdx.x; idx < 16 * DD; idx += blockDim.x) {
    int m = idx / DD, d = idx - m * DD;
    sq[m * 772 + d] = qem[idx];
  }
  __syncthreads();
  int wave = threadIdx.x >> 5;
  int lane = threadIdx.x & 31;
  int gwave = blockIdx.x * (blockDim.x >> 5) + wave;
  const int ntiles = EE / 16;  // 6250
  if (gwave >= ntiles) return;
  int e0 = gwave * 16;
  int half = lane >> 4, l16 = lane & 15;
  const float* brow = ent + (size_t)(e0 + l16) * DD;  // B[k][n] = ent[n][k]
  const float* arow = sq + l16 * 772;
  v8f c = {};
  for (int k0 = 0; k0 < DD; k0 += 4) {
    int ka = k0 + 2 * half;                 // lanes 0-15: K=k0,k0+1 ; 16-31: K=k0+2,k0+3
    v2f a = *(const v2f*)(arow + ka);       // ds_load_b64
    v2f bf = *(const v2f*)(brow + ka);      // global_load_b64
    c = __builtin_amdgcn_wmma_f32_16x16x4_f32(false, a, false, bf, (short)0, c,
                                              false, false);
  }
  const float scale = 0.03608439182435161f;  // 1/sqrt(768)
#pragma unroll
  for (int r = 0; r < 8; ++r) {
    int m = r + 8 * half;                    // C/D layout: VGPR r, lanes>=16 -> M+8
    float v = c[r] * scale;
    v = 1.f / (1.f + __expf(-v));
    direct[(size_t)m * EE + e0 + l16] = v;
  }
}

// -------------------- enhanced = heads*(1+0.3*direct); row sums via atomics
__global__ void k_enhance(const float* __restrict__ heads, const float* __restrict__ direct,
                          float* __restrict__ enh, float* __restrict__ red_sum) {
  int b = blockIdx.y;
  int e = blockIdx.x * blockDim.x + threadIdx.x;
  float v = 0.f;
  if (e < EE) {
    size_t idx = (size_t)b * EE + e;
    v = heads[idx] * (1.f + 0.3f * direct[idx]);
    enh[idx] = v;
  }
  __shared__ float sred[256];
  sred[threadIdx.x] = v;
  __syncthreads();
  for (int s = 128; s > 0; s >>= 1) {
    if (threadIdx.x < s) sred[threadIdx.x] += sred[threadIdx.x + s];
    __syncthreads();
  }
  if (threadIdx.x == 0) atomAddF(red_sum + b, sred[0]);
}

__global__ void k_rownorm(float* __restrict__ enh, const float* __restrict__ red_sum) {
  int b = blockIdx.y;
  int e = blockIdx.x * blockDim.x + threadIdx.x;
  if (e >= EE) return;
  float s = red_sum[b];
  float inv = (s > 0.f) ? 1.f / s : 1.f;
  enh[(size_t)b * EE + e] *= inv;
}

// ------------------------------- follow(): gather-multiply-scatter over edges
__global__ void k_scatter(const float* __restrict__ prev, const float* __restrict__ wr,
                          const int* __restrict__ subj, const int* __restrict__ rel,
                          const int* __restrict__ obj, float* __restrict__ nxt) {
  __shared__ float swr[BB * RR];
  for (int i = threadIdx.x; i < BB * RR; i += blockDim.x) swr[i] = wr[i];
  __syncthreads();
  int t = blockIdx.x * blockDim.x + threadIdx.x;
  if (t >= TT) return;
  int s = subj[t], r = rel[t], o = obj[t];
#pragma unroll
  for (int b = 0; b < BB; ++b) {
    float x = prev[(size_t)b * EE + s] * swr[b * RR + r];
    atomAddF(nxt + (size_t)b * EE + o, x);
  }
}

// clamp to [0,1] + per-row max (float-as-uint atomicMax valid for vals >= 0)
__global__ void k_clip_max(float* __restrict__ nxt, unsigned* __restrict__ red_max) {
  int b = blockIdx.y;
  int e = blockIdx.x * blockDim.x + threadIdx.x;
  float v = 0.f;
  if (e < EE) {
    size_t idx = (size_t)b * EE + e;
    v = fminf(fmaxf(nxt[idx], 0.f), 1.f);
    nxt[idx] = v;
  }
  __shared__ float sred[256];
  sred[threadIdx.x] = v;
  __syncthreads();
  for (int s = 128; s > 0; s >>= 1) {
    if (threadIdx.x < s) sred[threadIdx.x] = fmaxf(sred[threadIdx.x], sred[threadIdx.x + s]);
    __syncthreads();
  }
  if (threadIdx.x == 0) atomicMax(red_max + b, __float_as_uint(sred[0]));
}

// max-normalize, residual mix (t>0), *entity_range, accumulate row sums
__global__ void k_step_mix(float* __restrict__ nxt, const float* __restrict__ prev,
                           const float* __restrict__ erange,
                           const unsigned* __restrict__ red_max,
                           float* __restrict__ red_sum, int tstep) {
  int b = blockIdx.y;
  int e = blockIdx.x * blockDim.x + threadIdx.x;
  float m = __uint_as_float(red_max[b]);
  float minv = (m > 0.f) ? 1.f / m : 1.f;
  float v = 0.f;
  if (e < EE) {
    size_t idx = (size_t)b * EE + e;
    v = nxt[idx] * minv;
    if (tstep > 0) v = 0.7f * v + 0.3f * prev[idx];
    v *= erange[idx];
    nxt[idx] = v;
  }
  __shared__ float sred[256];
  sred[threadIdx.x] = v;
  __syncthreads();
  for (int s = 128; s > 0; s >>= 1) {
    if (threadIdx.x < s) sred[threadIdx.x] += sred[threadIdx.x + s];
    __syncthreads();
  }
  if (threadIdx.x == 0) atomAddF(red_sum + b, sred[0]);
}

// safe_row_norm -> last_e (cur), and accumulate hop-attention-weighted score
__global__ void k_norm_acc(const float* __restrict__ nxt, const float* __restrict__ red_sum,
                           float* __restrict__ cur, float* __restrict__ target,
                           const float* __restrict__ hop_attn, int w, int tstep) {
  int b = blockIdx.y;
  int e = blockIdx.x * blockDim.x + threadIdx.x;
  if (e >= EE) return;
  float s = red_sum[b];
  float inv = (s > 0.f) ? 1.f / s : 1.f;
  size_t idx = (size_t)b * EE + e;
  float le = nxt[idx] * inv;
  cur[idx] = le;
  float hw = hop_attn[(w * BB + b) * NSTEPS + tstep];
  target[idx] += hw * le;
}

// out += acc0 * (1 + 0.15*direct)   (way-0 final_direct_weight)
__global__ void k_combine(float* __restrict__ out, const float* __restrict__ acc0,
                          const float* __restrict__ direct) {
  int idx = blockIdx.x * blockDim.x + threadIdx.x;
  if (idx < BB * EE) out[idx] += acc0[idx] * (1.f + 0.15f * direct[idx]);
}

// ============================================================================
extern "C" void kernel_launch(void* const* d_in, const int* in_sizes, int n_in,
                              void* d_out, int out_size, void* d_ws, size_t ws_size,
                              hipStream_t stream) {
  (void)in_sizes; (void)n_in; (void)out_size; (void)ws_size;
  const float* heads  = (const float*)d_in[0];
  const float* qemb   = (const float*)d_in[1];
  const float* qwh    = (const float*)d_in[2];
  const int*   amask  = (const int*)d_in[3];
  const float* erange = (const float*)d_in[4];
  const int*   subj   = (const int*)d_in[5];
  const int*   rel    = (const int*)d_in[6];
  const int*   obj    = (const int*)d_in[7];
  const float* rimp   = (const float*)d_in[8];
  const float* ent    = (const float*)d_in[9];
  const float* mw     = (const float*)d_in[10];
  const float* mb     = (const float*)d_in[11];
  const float* sw     = (const float*)d_in[12];
  const float* sb     = (const float*)d_in[13];
  const float* rw     = (const float*)d_in[14];
  const float* rb     = (const float*)d_in[15];
  const float* hw     = (const float*)d_in[16];
  const float* hb     = (const float*)d_in[17];
  const float* temp   = (const float*)d_in[18];
  float* out = (float*)d_out;

  float* ws = (float*)d_ws;
  float*    qem      = ws;                                   // 16*768
  float*    cq_all   = qem + BB * DD;                        // 6*16*768
  float*    wr_all   = cq_all + 6 * BB * DD;                 // 6*16*200
  float*    hop_attn = wr_all + 6 * BB * RR;                 // 2*16*3
  float*    red      = ws + 105312;                          // 64 floats
  float*    red_sum  = red;
  unsigned* red_max  = (unsigned*)(red + 16);
  float*    direct   = ws + 105472;                          // [B,E]
  float*    enh      = direct + BB * EE;                     // [B,E]
  float*    cur      = enh + BB * EE;                        // [B,E]
  float*    nxt      = cur + BB * EE;                        // [B,E]
  float*    acc0     = nxt + BB * EE;                        // [B,E]

  const int beBlocks = (BB * EE + 255) / 256;                // 6250
  dim3 rowGrid((EE + 255) / 256, BB);                        // (391, 16)

  k_init<<<beBlocks, 256, 0, stream>>>(out, acc0, red);
  k_qem<<<(DD + 127) / 128, 128, 0, stream>>>(qemb, mw, mb, qem);
  k_cq<<<dim3((DD + 127) / 128, 6), 128, 0, stream>>>(qemb, sw, sb, cq_all);
  k_hop<<<1, 32, 0, stream>>>(qemb, hw, hb, hop_attn);
  k_ctx_rel<<<dim3(6, BB), 256, 0, stream>>>(cq_all, qwh, amask, rw, rb, rimp, temp, wr_all);

  const int tiles = EE / 16;                                 // 6250
  k_direct<<<(tiles + 3) / 4, 128, 0, stream>>>(qem, ent, direct);

  k_enhance<<<rowGrid, 256, 0, stream>>>(heads, direct, enh, red_sum);
  k_rownorm<<<rowGrid, 256, 0, stream>>>(enh, red_sum);

  for (int w = 0; w < NWAYS; ++w) {
    const float* prev = enh;
    for (int t = 0; t < NSTEPS; ++t) {
      int i = w * NSTEPS + t;
      k_fzero<<<beBlocks, 256, 0, stream>>>(nxt, red);
      k_scatter<<<(TT + 255) / 256, 256, 0, stream>>>(prev, wr_all + (size_t)i * BB * RR,
                                                      subj, rel, obj, nxt);
      k_clip_max<<<rowGrid, 256, 0, stream>>>(nxt, red_max);
      k_step_mix<<<rowGrid, 256, 0, stream>>>(nxt, prev, erange, red_max, red_sum, t);
      float* target = (w == 0) ? acc0 : out;
      k_norm_acc<<<rowGrid, 256, 0, stream>>>(nxt, red_sum, cur, target, hop_attn, w, t);
      prev = cur;
    }
  }
  k_combine<<<beBlocks, 256, 0, stream>>>(out, acc0, direct);
}